// WorldSSM_37014028157386
// MI455X (gfx1250) — compile-verified
//
#include <hip/hip_runtime.h>

// Problem dims
#define D_MODEL 1024
#define D_STATE 256
#define BATCH   8
#define SEQ     2048
#define CHUNK   32
#define NCHUNK  (SEQ / CHUNK)

typedef float v2f __attribute__((ext_vector_type(2)));
typedef float v8f __attribute__((ext_vector_type(8)));

__device__ __forceinline__ v8f wmma_f32(v2f a, v2f b, v8f c) {
  // D = A(16x4) * B(4x16) + C(16x16), f32 exact
  return __builtin_amdgcn_wmma_f32_16x16x4_f32(false, a, false, b,
                                               (short)0, c, false, false);
}

// ---------------------------------------------------------------------------
// Dense WMMA GEMM:  Out = alpha*(A @ B) [+ I] [+ epx .* epD]
// A: MxK row-major, B: KxN row-major, Out: MxN.  M%64==0, N%128==0, K%32==0.
// 256 threads = 8 wave32s in a 2x4 grid; each wave computes a 32x32 register
// tile (2x2 WMMA tiles) -> 4 WMMAs per A/B fragment pair, halving LDS traffic
// per WMMA and providing 4 independent accumulation chains.
// ---------------------------------------------------------------------------
__global__ __launch_bounds__(256) void dense_gemm_wmma(
    const float* __restrict__ A, const float* __restrict__ Bm,
    float* __restrict__ Out, int M, int N, int K,
    float alpha, int addI,
    const float* __restrict__ epx, const float* __restrict__ epD) {
  __shared__ float sA[64][36];    // 64x32 panel, padded row (conflict-free)
  __shared__ float sB[32][132];   // 32x128 panel, padded row

  const int tid  = threadIdx.x;
  const int lane = tid & 31;
  const int wave = tid >> 5;
  const int wmB = (wave >> 2) * 32;    // 0 / 32
  const int wnB = (wave & 3) * 32;     // 0..96
  const int half = lane >> 4;
  const int lrow = lane & 15;
  const long m0 = (long)blockIdx.x * 64;
  const long n0 = (long)blockIdx.y * 128;

  // A panel: 64x32 = 512 float4, 2 per thread
  const int ra0 = tid >> 3,          ca = (tid & 7) << 2;
  const int ra1 = ra0 + 32;
  // B panel: 32x128 = 1024 float4, 4 per thread
  const int rb0 = tid >> 5,          cb = (tid & 31) << 2;  // +8 rows per pass

  v8f acc[2][2] = {{{}, {}}, {{}, {}}};

  for (int k0 = 0; k0 < K; k0 += 32) {
    if (k0 + 32 < K) {
      __builtin_prefetch(A + (m0 + ra0) * (long)K + (k0 + 32) + ca, 0, 1);
      __builtin_prefetch(Bm + (long)(k0 + 32 + rb0) * N + n0 + cb, 0, 1);
    }
    float4 va0 = *(const float4*)(A + (m0 + ra0) * (long)K + k0 + ca);
    float4 va1 = *(const float4*)(A + (m0 + ra1) * (long)K + k0 + ca);
    float4 vb0 = *(const float4*)(Bm + (long)(k0 + rb0     ) * N + n0 + cb);
    float4 vb1 = *(const float4*)(Bm + (long)(k0 + rb0 +  8) * N + n0 + cb);
    float4 vb2 = *(const float4*)(Bm + (long)(k0 + rb0 + 16) * N + n0 + cb);
    float4 vb3 = *(const float4*)(Bm + (long)(k0 + rb0 + 24) * N + n0 + cb);
    __syncthreads();
    *(float4*)&sA[ra0][ca]      = va0;
    *(float4*)&sA[ra1][ca]      = va1;
    *(float4*)&sB[rb0][cb]      = vb0;
    *(float4*)&sB[rb0 +  8][cb] = vb1;
    *(float4*)&sB[rb0 + 16][cb] = vb2;
    *(float4*)&sB[rb0 + 24][cb] = vb3;
    __syncthreads();
#pragma unroll
    for (int ks = 0; ks < 8; ++ks) {
      const int k2 = ks * 4 + half * 2;
      v2f a0, a1, b0, b1;
      a0.x = sA[wmB + lrow][k2];        a0.y = sA[wmB + lrow][k2 + 1];
      a1.x = sA[wmB + 16 + lrow][k2];   a1.y = sA[wmB + 16 + lrow][k2 + 1];
      b0.x = sB[k2][wnB + lrow];        b0.y = sB[k2 + 1][wnB + lrow];
      b1.x = sB[k2][wnB + 16 + lrow];   b1.y = sB[k2 + 1][wnB + 16 + lrow];
      acc[0][0] = wmma_f32(a0, b0, acc[0][0]);
      acc[0][1] = wmma_f32(a0, b1, acc[0][1]);
      acc[1][0] = wmma_f32(a1, b0, acc[1][0]);
      acc[1][1] = wmma_f32(a1, b1, acc[1][1]);
    }
  }

#pragma unroll
  for (int ti = 0; ti < 2; ++ti)
#pragma unroll
    for (int tj = 0; tj < 2; ++tj) {
      const long gcol = n0 + wnB + tj * 16 + lrow;
#pragma unroll
      for (int r = 0; r < 8; ++r) {
        const long grow = m0 + wmB + ti * 16 + half * 8 + r;
        float v = alpha * acc[ti][tj][r];
        if (addI && grow == gcol) v += 1.0f;
        if (epx) v += epx[grow * N + gcol] * epD[gcol];
        Out[grow * N + gcol] = v;
      }
    }
}

// ---------------------------------------------------------------------------
// Mapped/accumulating WMMA GEMM for the chunked scan.  K = N = 256.
//   mode 0 (conv, shift k): per chunk c: V[b, c*L + j + k, :] += u[b, c*L + j, :] @ P
//       virtual rows vr enumerate (c, j, b); Mvirt = NCHUNK*8*(L-k)
//   mode 1 (reconstruct, i=k): h[b, c*L + i, :] += Hcarry[c*8+b, :] @ P
//       vr enumerates (c, b); Mvirt = NCHUNK*8
// ---------------------------------------------------------------------------
__device__ __forceinline__ long map_src_row(int vr, int mode, int kk) {
  if (mode == 0) {
    const int Rc = 8 * (CHUNK - kk);
    const int c = vr / Rc, rem = vr - c * Rc;
    const int j = rem >> 3, b = rem & 7;
    return (long)b * SEQ + c * CHUNK + j;
  }
  return vr;  // Hcarry is contiguous [(c*8+b), 256]
}
__device__ __forceinline__ long map_out_row(int vr, int mode, int kk) {
  if (mode == 0) {
    const int Rc = 8 * (CHUNK - kk);
    const int c = vr / Rc, rem = vr - c * Rc;
    const int j = rem >> 3, b = rem & 7;
    return (long)b * SEQ + c * CHUNK + j + kk;
  }
  const int c = vr >> 3, b = vr & 7;
  return (long)b * SEQ + c * CHUNK + kk;
}

__global__ __launch_bounds__(256) void map_gemm_wmma(
    const float* __restrict__ Asrc, const float* __restrict__ P,
    float* __restrict__ Out, int mode, int kk) {
  __shared__ float sA[64][36];
  __shared__ float sB[32][132];

  const int tid  = threadIdx.x;
  const int lane = tid & 31;
  const int wave = tid >> 5;
  const int wmB = (wave >> 2) * 32;
  const int wnB = (wave & 3) * 32;
  const int half = lane >> 4;
  const int lrow = lane & 15;
  const int m0 = blockIdx.x * 64;
  const long n0 = (long)blockIdx.y * 128;

  const int ra0 = tid >> 3, ca = (tid & 7) << 2;
  const int ra1 = ra0 + 32;
  const int rb0 = tid >> 5, cb = (tid & 31) << 2;
  const long arow0 = map_src_row(m0 + ra0, mode, kk) * 256;
  const long arow1 = map_src_row(m0 + ra1, mode, kk) * 256;

  v8f acc[2][2] = {{{}, {}}, {{}, {}}};

  for (int k0 = 0; k0 < 256; k0 += 32) {
    float4 va0 = *(const float4*)(Asrc + arow0 + k0 + ca);
    float4 va1 = *(const float4*)(Asrc + arow1 + k0 + ca);
    float4 vb0 = *(const float4*)(P + (long)(k0 + rb0     ) * 256 + n0 + cb);
    float4 vb1 = *(const float4*)(P + (long)(k0 + rb0 +  8) * 256 + n0 + cb);
    float4 vb2 = *(const float4*)(P + (long)(k0 + rb0 + 16) * 256 + n0 + cb);
    float4 vb3 = *(const float4*)(P + (long)(k0 + rb0 + 24) * 256 + n0 + cb);
    __syncthreads();
    *(float4*)&sA[ra0][ca]      = va0;
    *(float4*)&sA[ra1][ca]      = va1;
    *(float4*)&sB[rb0][cb]      = vb0;
    *(float4*)&sB[rb0 +  8][cb] = vb1;
    *(float4*)&sB[rb0 + 16][cb] = vb2;
    *(float4*)&sB[rb0 + 24][cb] = vb3;
    __syncthreads();
#pragma unroll
    for (int ks = 0; ks < 8; ++ks) {
      const int k2 = ks * 4 + half * 2;
      v2f a0, a1, b0, b1;
      a0.x = sA[wmB + lrow][k2];        a0.y = sA[wmB + lrow][k2 + 1];
      a1.x = sA[wmB + 16 + lrow][k2];   a1.y = sA[wmB + 16 + lrow][k2 + 1];
      b0.x = sB[k2][wnB + lrow];        b0.y = sB[k2 + 1][wnB + lrow];
      b1.x = sB[k2][wnB + 16 + lrow];   b1.y = sB[k2 + 1][wnB + 16 + lrow];
      acc[0][0] = wmma_f32(a0, b0, acc[0][0]);
      acc[0][1] = wmma_f32(a0, b1, acc[0][1]);
      acc[1][0] = wmma_f32(a1, b0, acc[1][0]);
      acc[1][1] = wmma_f32(a1, b1, acc[1][1]);
    }
  }

#pragma unroll
  for (int ti = 0; ti < 2; ++ti)
#pragma unroll
    for (int tj = 0; tj < 2; ++tj) {
      const long gcol = n0 + wnB + tj * 16 + lrow;
#pragma unroll
      for (int r = 0; r < 8; ++r) {
        const int vr = m0 + wmB + ti * 16 + half * 8 + r;
        const long orow = map_out_row(vr, mode, kk) * 256;
        Out[orow + gcol] += acc[ti][tj][r];
      }
    }
}

// ---------------------------------------------------------------------------
// Sequential carry scan over chunk boundaries: 64 steps instead of 2048.
// One workgroup; A^L (256 KB) held in the WGP's 320 KB LDS.
//   H_{c+1} = H_c @ P_L + V[:, c*L + L-1, :],  Hcarry[c] = H_c
// ---------------------------------------------------------------------------
__global__ __launch_bounds__(256) void carry_scan(
    const float* __restrict__ h0, const float* __restrict__ V,
    const float* __restrict__ PL, float* __restrict__ Hc) {
  extern __shared__ float smem[];
  float* sP = smem;                 // 256*256
  float* sH = smem + 256 * 256;     // 8*256
  const int n = threadIdx.x;

  for (int r = 0; r < 256; ++r) sP[r * 256 + n] = PL[r * 256 + n];

  float h[8];
#pragma unroll
  for (int b = 0; b < 8; ++b) h[b] = h0[b * 256 + n];

  for (int c = 0; c < NCHUNK; ++c) {
#pragma unroll
    for (int b = 0; b < 8; ++b) {
      Hc[((long)c * 8 + b) * 256 + n] = h[b];
      sH[b * 256 + n] = h[b];
    }
    __syncthreads();
    float acc[8];
#pragma unroll
    for (int b = 0; b < 8; ++b)
      acc[b] = V[((long)b * SEQ + c * CHUNK + CHUNK - 1) * 256 + n];
    for (int m = 0; m < 256; ++m) {
      const float p = sP[m * 256 + n];
#pragma unroll
      for (int b = 0; b < 8; ++b) acc[b] = fmaf(sH[b * 256 + m], p, acc[b]);
    }
    __syncthreads();
#pragma unroll
    for (int b = 0; b < 8; ++b) h[b] = acc[b];
  }
}

__global__ void ident256(float* __restrict__ T) {
  const int r = blockIdx.x, c = threadIdx.x;
  T[r * 256 + c] = (r == c) ? 1.0f : 0.0f;
}

// ---------------------------------------------------------------------------
extern "C" void kernel_launch(void* const* d_in, const int* in_sizes, int n_in,
                              void* d_out, int out_size, void* d_ws, size_t ws_size,
                              hipStream_t stream) {
  (void)in_sizes; (void)n_in; (void)out_size; (void)ws_size;
  const float* x  = (const float*)d_in[0];  // (B, S, D_MODEL)
  const float* A  = (const float*)d_in[1];  // (256, 256)
  const float* Bp = (const float*)d_in[2];  // (D_MODEL, 256)
  const float* Cp = (const float*)d_in[3];  // (256, D_MODEL)
  const float* Dp = (const float*)d_in[4];  // (D_MODEL,)
  const float* h0 = (const float*)d_in[5];  // (B, 256)
  float* y = (float*)d_out;

  float* ws = (float*)d_ws;
  float* T0 = ws;
  float* T1 = ws + 65536;
  float* P  = ws + 2 * 65536;                    // slots 0..CHUNK; P[k] at +k*65536
  float* U  = P + (CHUNK + 1) * 65536;           // (B*S, 256)
  float* V  = U + (long)BATCH * SEQ * 256;       // (B*S, 256) -> becomes h
  float* Hc = V + (long)BATCH * SEQ * 256;       // (NCHUNK*8, 256)

  const dim3 blk(256);
  const dim3 g256(256 / 64, 256 / 128);          // 256x256 GEMMs

  // 1) A_d = expm(0.1*A) via Horner Taylor (order 12):  T <- (0.1/j)*(A@T) + I
  ident256<<<256, 256, 0, stream>>>(T0);
  float* tc = T0; float* tn = T1;
  for (int j = 12; j >= 2; --j) {
    dense_gemm_wmma<<<g256, blk, 0, stream>>>(
        A, tc, tn, 256, 256, 256, 0.1f / (float)j, 1, nullptr, nullptr);
    float* t = tc; tc = tn; tn = t;
  }
  dense_gemm_wmma<<<g256, blk, 0, stream>>>(
      A, tc, P + 65536, 256, 256, 256, 0.1f, 1, nullptr, nullptr);  // P[1]=A_d

  // 2) Powers P[k] = P[k-1] @ A_d, k = 2..CHUNK
  for (int k = 2; k <= CHUNK; ++k)
    dense_gemm_wmma<<<g256, blk, 0, stream>>>(
        P + (long)(k - 1) * 65536, P + 65536, P + (long)k * 65536,
        256, 256, 256, 1.0f, 0, nullptr, nullptr);

  // 3) u = x @ B   (16384 x 256 x 1024)
  dense_gemm_wmma<<<dim3(16384 / 64, 256 / 128), blk, 0, stream>>>(
      x, Bp, U, 16384, 256, 1024, 1.0f, 0, nullptr, nullptr);

  // 4) V = u (the k=0 term, P[0] = I)
  hipMemcpyAsync(V, U, (size_t)BATCH * SEQ * 256 * sizeof(float),
                 hipMemcpyDeviceToDevice, stream);

  // 5) Within-chunk convolution: V[:, j+k] += u[:, j] @ P[k], k = 1..CHUNK-1
  for (int k = 1; k < CHUNK; ++k) {
    const int Mv = NCHUNK * 8 * (CHUNK - k);
    map_gemm_wmma<<<dim3(Mv / 64, 2), blk, 0, stream>>>(U, P + (long)k * 65536, V, 0, k);
  }

  // 6) Sequential carry across chunks (64 steps, A^L in LDS)
  carry_scan<<<1, blk, (256 * 256 + 8 * 256) * sizeof(float), stream>>>(
      h0, V, P + (long)CHUNK * 65536, Hc);

  // 7) Reconstruct: h[:, c*L+i] = Hcarry[c] @ P[i+1] + V[:, c*L+i]
  for (int i = 0; i < CHUNK; ++i)
    map_gemm_wmma<<<dim3((NCHUNK * 8) / 64, 2), blk, 0, stream>>>(
        Hc, P + (long)(i + 1) * 65536, V, 1, i);

  // 8) y = h @ C + x * D   (16384 x 1024 x 256, fused epilogue)
  dense_gemm_wmma<<<dim3(16384 / 64, 1024 / 128), blk, 0, stream>>>(
      V, Cp, y, 16384, 1024, 256, 1.0f, 0, x, Dp);
}